// NegativeLogLikelihood_781684048657
// MI455X (gfx1250) — compile-verified
//
#include <hip/hip_runtime.h>
#include <hip/hip_bf16.h>
#include <math.h>

#define L2_REG 0.0001f

typedef __attribute__((ext_vector_type(2))) float v2f;
typedef __attribute__((ext_vector_type(4))) float v4f;
typedef __attribute__((ext_vector_type(8))) float v8f;

// ---------------------------------------------------------------------------
// Kernel 1: exp(r) precompute + zero the 3 global accumulators.
// ---------------------------------------------------------------------------
__global__ void cox_init_kernel(const float* __restrict__ r,
                                float* __restrict__ exp_r,
                                float* __restrict__ accums, int N) {
    int i = blockIdx.x * blockDim.x + threadIdx.x;
    if (i < 3) accums[i] = 0.0f;          // [0]=sum_loss [1]=e_sum [2]=sumsq_W
    if (i < N) exp_r[i] = expf(r[i]);
}

// ---------------------------------------------------------------------------
// Kernel 2: num/den via V_WMMA_F32_16X16X4_F32.
//   D = A(16x4 mask) x B(4x16 [exp_r | 1 | 0...]) + C
//   C column 0 accumulates num, column 1 accumulates den (exact f32).
// One block (8 wave32) per 16-row tile; waves split the K range.
// Inner loop: uniform scalar bounds (readfirstlane), b128 loads, 4 WMMAs/trip,
// branch-free A/B construction (EXEC stays all-ones around every WMMA).
// ---------------------------------------------------------------------------
__global__ __launch_bounds__(256) void cox_wmma_kernel(
    const float* __restrict__ y, const float* __restrict__ exp_r,
    float* __restrict__ num, float* __restrict__ den, int N) {

    __shared__ float s_num[16];
    __shared__ float s_den[16];
    const int tid = threadIdx.x;
    if (tid < 16) { s_num[tid] = 0.0f; s_den[tid] = 0.0f; }
    __syncthreads();

    const int wave = __builtin_amdgcn_readfirstlane(tid >> 5); // uniform 0..7
    const int lane = tid & 31;
    const int kh   = lane >> 4;         // lane half: 0 -> K{0,1}, 1 -> K{2,3}
    const int l16  = lane & 15;

    const int i0 = blockIdx.x * 16;     // row-tile base
    const float y_row = y[i0 + l16];    // A layout: both halves hold M = lane%16

    // B-column selectors as arithmetic (no divergent control flow):
    // B[k, n] = n==0 ? exp_r[j0+k] : (n==1 ? 1 : 0)  ==  fma(er, is0, is1)
    const float is0 = (l16 == 0) ? 1.0f : 0.0f;
    const float is1 = (l16 == 1) ? 1.0f : 0.0f;

    const int kchunk = N >> 3;          // N/8 columns per wave (1536)
    const int kbeg = wave * kchunk;     // uniform
    const int kend = kbeg + kchunk;     // uniform

    v8f c = {};
    for (int j0 = kbeg; j0 < kend; j0 += 16) {
        #pragma unroll
        for (int u = 0; u < 4; ++u) {
            const int j = j0 + 4 * u;
            // Uniform-address 16B loads of the shared K-slice.
            const v4f y4  = *(const v4f*)(y + j);
            const v4f er4 = *(const v4f*)(exp_r + j);
            // Lane half picks its K pair (K{0,1} vs K{2,3}).
            const float yj0 = kh ? y4.z  : y4.x;
            const float yj1 = kh ? y4.w  : y4.y;
            const float er0 = kh ? er4.z : er4.x;
            const float er1 = kh ? er4.w : er4.y;

            v2f a, b;
            // A: mask[i,j] = (y[i] - y[j] <= 0) -> exact 0/1 in f32
            a.x = (y_row <= yj0) ? 1.0f : 0.0f;
            a.y = (y_row <= yj1) ? 1.0f : 0.0f;
            // B: branch-free column build (er finite, so er*0 == 0 exactly)
            b.x = fmaf(er0, is0, is1);
            b.y = fmaf(er1, is0, is1);

            c = __builtin_amdgcn_wmma_f32_16x16x4_f32(
                    /*neg_a=*/false, a, /*neg_b=*/false, b,
                    /*c_mod=*/(short)0, c, /*reuse_a=*/false, /*reuse_b=*/false);
        }
    }

    // C layout: VGPR g -> lanes 0-15: (M=g, N=lane); lanes 16-31: (M=g+8, N=lane-16)
    // Columns 0 (num) and 1 (den) live in lanes {0,16} and {1,17}.
    if (l16 < 2) {
        float* dst = (l16 == 0) ? s_num : s_den;
        #pragma unroll
        for (int g = 0; g < 8; ++g)
            atomicAdd(&dst[8 * kh + g], c[g]);
    }
    __syncthreads();
    if (tid < 16) {                     // this block exclusively owns the row tile
        num[i0 + tid] = s_num[tid];
        den[i0 + tid] = s_den[tid];
    }
}

// ---------------------------------------------------------------------------
// Kernel 3: sum of squares of W (Frobenius norm^2) -> accums[2].
// ---------------------------------------------------------------------------
__global__ __launch_bounds__(256) void wnorm_kernel(const float* __restrict__ W,
                                                    float* __restrict__ accums, int M) {
    __shared__ float s[256];
    float acc = 0.0f;
    for (int i = blockIdx.x * blockDim.x + threadIdx.x; i < M;
         i += gridDim.x * blockDim.x) {
        float w = W[i];
        acc += w * w;
    }
    s[threadIdx.x] = acc;
    __syncthreads();
    for (int sft = 128; sft > 0; sft >>= 1) {
        if (threadIdx.x < sft) s[threadIdx.x] += s[threadIdx.x + sft];
        __syncthreads();
    }
    if (threadIdx.x == 0) atomicAdd(&accums[2], s[0]);
}

// ---------------------------------------------------------------------------
// Kernel 4: per-sample loss contribution + e_sum, block-reduced.
// ---------------------------------------------------------------------------
__global__ __launch_bounds__(256) void loss_reduce_kernel(
    const float* __restrict__ r, const int* __restrict__ ev,
    const float* __restrict__ num, const float* __restrict__ den,
    float* __restrict__ accums, int N) {

    __shared__ float s_loss[256];
    __shared__ float s_e[256];
    const int i = blockIdx.x * blockDim.x + threadIdx.x;
    float li = 0.0f, ei = 0.0f;
    if (i < N) {
        ei = (float)ev[i];
        float log_loss = logf(num[i] / den[i]);
        li = (r[i] - log_loss) * ei;
    }
    s_loss[threadIdx.x] = li;
    s_e[threadIdx.x] = ei;
    __syncthreads();
    for (int sft = 128; sft > 0; sft >>= 1) {
        if (threadIdx.x < sft) {
            s_loss[threadIdx.x] += s_loss[threadIdx.x + sft];
            s_e[threadIdx.x]    += s_e[threadIdx.x + sft];
        }
        __syncthreads();
    }
    if (threadIdx.x == 0) {
        atomicAdd(&accums[0], s_loss[0]);
        atomicAdd(&accums[1], s_e[0]);
    }
}

// ---------------------------------------------------------------------------
// Kernel 5: final scalar.
// ---------------------------------------------------------------------------
__global__ void final_kernel(const float* __restrict__ accums,
                             float* __restrict__ out) {
    if (threadIdx.x == 0 && blockIdx.x == 0) {
        float sum_loss = accums[0];
        float e_sum    = accums[1];
        float sumsq    = accums[2];
        float loss = -sum_loss / e_sum + L2_REG * sqrtf(sumsq);
        out[0] = (e_sum == 0.0f) ? 0.0f : loss;
    }
}

// ---------------------------------------------------------------------------
extern "C" void kernel_launch(void* const* d_in, const int* in_sizes, int n_in,
                              void* d_out, int out_size, void* d_ws, size_t ws_size,
                              hipStream_t stream) {
    const float* r = (const float*)d_in[0];   // risk_pred (N)
    const float* y = (const float*)d_in[1];   // y (N)
    const int*   e = (const int*)d_in[2];     // e (N)
    const float* W = (const float*)d_in[3];   // W (1024*1024)
    const int N  = in_sizes[0];               // 12288 (divisible by 16 and 128)
    const int MW = in_sizes[3];

    float* ws     = (float*)d_ws;
    float* exp_r  = ws;                       // N
    float* num    = ws + N;                   // N
    float* den    = ws + 2 * N;               // N
    float* accums = ws + 3 * N;               // 3

    cox_init_kernel<<<(N + 255) / 256, 256, 0, stream>>>(r, exp_r, accums, N);
    cox_wmma_kernel<<<N / 16, 256, 0, stream>>>(y, exp_r, num, den, N);
    wnorm_kernel<<<256, 256, 0, stream>>>(W, accums, MW);
    loss_reduce_kernel<<<(N + 255) / 256, 256, 0, stream>>>(r, e, num, den, accums, N);
    final_kernel<<<1, 1, 0, stream>>>(accums, (float*)d_out);
}